// GCN_39539468926991
// MI455X (gfx1250) — compile-verified
//
#include <hip/hip_runtime.h>
#include <hip/hip_bf16.h>

// ---------------------------------------------------------------------------
// GCN forward on MI455X (gfx1250, wave32).
//   X (= d_out) : [N,64] node state, updated in place per layer
//   d_ws        : M = messages [N,64] ; H = scatter accum [N,64]
// GEMMs use V_WMMA_F32_16X16X4_F32 (f32 matrix pipe, keeps fp32 semantics;
// compute is ~10.6 GFLOP total -> nowhere near any ceiling, memory/atomics
// dominate and the hot set (~141MB) is L2-resident on the 192MB L2).
// ---------------------------------------------------------------------------

typedef float v2f __attribute__((ext_vector_type(2)));
typedef float v8f __attribute__((ext_vector_type(8)));

#define NUSR 50000
#define NITM 50000
#define NTOT 100000
#define NEDG 1600000

__device__ __forceinline__ float lrelu(float v) { return v >= 0.0f ? v : 0.01f * v; }

// ------------------------- simple utility kernels --------------------------

__global__ __launch_bounds__(256) void copy4_kernel(const float4* __restrict__ s,
                                                    float4* __restrict__ d, int n4) {
  int i = blockIdx.x * blockDim.x + threadIdx.x;
  if (i < n4) d[i] = s[i];
}

__global__ __launch_bounds__(256) void zero4_kernel(float4* __restrict__ d, int n4) {
  int i = blockIdx.x * blockDim.x + threadIdx.x;
  if (i < n4) d[i] = make_float4(0.f, 0.f, 0.f, 0.f);
}

// Row-wise L2 normalize: one wave (32 lanes) per row of 64 floats.
__global__ __launch_bounds__(256) void normalize_kernel(float* __restrict__ X, int nrows) {
  const int lane = threadIdx.x & 31;
  const int row  = (blockIdx.x * blockDim.x + threadIdx.x) >> 5;
  if (row >= nrows) return;
  float* p = X + (size_t)row * 64 + 2 * lane;
  v2f v = *(const v2f*)p;
  float ss = v.x * v.x + v.y * v.y;
  #pragma unroll
  for (int off = 16; off > 0; off >>= 1) ss += __shfl_xor(ss, off, 32);
  float inv = 1.0f / fmaxf(sqrtf(ss), 1e-12f);
  v.x *= inv; v.y *= inv;
  *(v2f*)p = v;
}

// ------------------------- WMMA f32 GEMM: Y = X @ W^T + b ------------------
// X: [nrows, K], W: [64, K] row-major, b: [64], Y: [nrows, 64].
// One wave handles a 16-row tile; A fragments preloaded, 4 N-tiles of 16.
// A 16x4 f32 layout (ISA 7.12.2): lanes 0-15 hold K=k0+{0,1}, lanes 16-31
// hold K=k0+{2,3}.  C/D: vgpr g, lane l -> row g + 8*(l>>4), col l&15.

template <int K>
__global__ __launch_bounds__(256)
void gemm_bias_wmma(const float* __restrict__ X, const float* __restrict__ W,
                    const float* __restrict__ Bv, float* __restrict__ Y, int nrows) {
  constexpr int KS = K / 4;
  const int lane = threadIdx.x & 31;
  const int wid  = (blockIdx.x * blockDim.x + threadIdx.x) >> 5;
  const int row0 = wid * 16;
  if (row0 >= nrows) return;                 // wave-uniform: EXEC stays all-1s
  const int m  = lane & 15;
  const int hi = lane >> 4;

  v2f a[KS];
  const float* xr = X + (size_t)(row0 + m) * K + 2 * hi;
  #pragma unroll
  for (int s = 0; s < KS; ++s) a[s] = *(const v2f*)(xr + 4 * s);

  #pragma unroll
  for (int nt = 0; nt < 4; ++nt) {
    const int n = nt * 16 + m;
    const float bias = Bv[n];
    v8f c = {bias, bias, bias, bias, bias, bias, bias, bias};
    const float* wr = W + (size_t)n * K + 2 * hi;
    #pragma unroll
    for (int s = 0; s < KS; ++s) {
      v2f b = *(const v2f*)(wr + 4 * s);
      c = __builtin_amdgcn_wmma_f32_16x16x4_f32(false, a[s], false, b,
                                                (short)0, c, false, false);
    }
    #pragma unroll
    for (int g = 0; g < 8; ++g) {
      const int r = row0 + g + 8 * hi;
      if (r < nrows) Y[(size_t)r * 64 + n] = c[g];
    }
  }
}

// ------------------------- edge scatter-add --------------------------------
// h[dst] += m[src]; 16 lanes per edge, float4 gather + 4 f32 atomics each.
// H fits in L2 (25.6MB) so atomics resolve at the L2 atomic units.

__global__ __launch_bounds__(256)
void scatter_kernel(const int* __restrict__ esrc, const int* __restrict__ edst,
                    const float* __restrict__ M, float* __restrict__ H, int ne) {
  int t = blockIdx.x * blockDim.x + threadIdx.x;
  int e = t >> 4;
  if (e >= ne) return;
  int q = (t & 15) * 4;
  int s = esrc[e], d = edst[e];
  float4 v = *(const float4*)(M + (size_t)s * 64 + q);
  float* hp = H + (size_t)d * 64 + q;
  atomicAdd(hp + 0, v.x);
  atomicAdd(hp + 1, v.y);
  atomicAdd(hp + 2, v.z);
  atomicAdd(hp + 3, v.w);
}

// ------------------------- fused layer tail --------------------------------
// x_new = lrelu( lrelu(H) @ g_w^T + g_b + lrelu(x @ lin_w^T + lin_b) + id_emb )
// In place on X: each 16-row tile is private to its wave.

__global__ __launch_bounds__(256)
void combine_kernel(float* __restrict__ X, const float* __restrict__ H,
                    const float* __restrict__ IDE,
                    const float* __restrict__ LW, const float* __restrict__ LB,
                    const float* __restrict__ GW, const float* __restrict__ GB,
                    int nrows) {
  const int lane = threadIdx.x & 31;
  const int wid  = (blockIdx.x * blockDim.x + threadIdx.x) >> 5;
  const int row0 = wid * 16;
  if (row0 >= nrows) return;
  const int m  = lane & 15;
  const int hi = lane >> 4;

  v2f ax[16], ah[16];
  const float* xr = X + (size_t)(row0 + m) * 64 + 2 * hi;
  const float* hr = H + (size_t)(row0 + m) * 64 + 2 * hi;
  #pragma unroll
  for (int s = 0; s < 16; ++s) {
    ax[s] = *(const v2f*)(xr + 4 * s);
    v2f t = *(const v2f*)(hr + 4 * s);
    t.x = lrelu(t.x); t.y = lrelu(t.y);
    ah[s] = t;
  }

  #pragma unroll
  for (int nt = 0; nt < 4; ++nt) {
    const int n = nt * 16 + m;
    const float lb = LB[n], gb = GB[n];
    v8f cu = {lb, lb, lb, lb, lb, lb, lb, lb};
    v8f cg = {gb, gb, gb, gb, gb, gb, gb, gb};
    const float* lw = LW + (size_t)n * 64 + 2 * hi;
    const float* gw = GW + (size_t)n * 64 + 2 * hi;
    #pragma unroll
    for (int s = 0; s < 16; ++s) {
      v2f bl = *(const v2f*)(lw + 4 * s);
      v2f bg = *(const v2f*)(gw + 4 * s);
      cu = __builtin_amdgcn_wmma_f32_16x16x4_f32(false, ax[s], false, bl,
                                                 (short)0, cu, false, false);
      cg = __builtin_amdgcn_wmma_f32_16x16x4_f32(false, ah[s], false, bg,
                                                 (short)0, cg, false, false);
    }
    #pragma unroll
    for (int g = 0; g < 8; ++g) {
      const int r = row0 + g + 8 * hi;
      if (r < nrows) {
        float u = lrelu(cu[g]) + IDE[(size_t)r * 64 + n];
        X[(size_t)r * 64 + n] = lrelu(cg[g] + u);
      }
    }
  }
}

// ------------------------- launcher ----------------------------------------

extern "C" void kernel_launch(void* const* d_in, const int* in_sizes, int n_in,
                              void* d_out, int out_size, void* d_ws, size_t ws_size,
                              hipStream_t stream) {
  const float* features = (const float*)d_in[0];   // [NITM,128]
  const float* id_emb   = (const float*)d_in[1];   // [NTOT,64]
  const int*   edges    = (const int*)d_in[2];     // [2,NEDG]
  const float* pref     = (const float*)d_in[3];   // [NUSR,64]
  const float* mlp_w    = (const float*)d_in[4];   // [64,128]
  const float* mlp_b    = (const float*)d_in[5];   // [64]

  float* X = (float*)d_out;                        // [NTOT,64] node state
  float* M = (float*)d_ws;                         // [NTOT,64] messages
  float* H = M + (size_t)NTOT * 64;                // [NTOT,64] scatter accum

  const int BLK = 256;
  // wave-per-16-rows grids
  const int gemm_blocks_N  = ((NTOT + 15) / 16 + 7) / 8;
  const int gemm_blocks_NI = ((NITM + 15) / 16 + 7) / 8;

  // 1) X[0:NU] = pref
  {
    int n4 = NUSR * 64 / 4;
    copy4_kernel<<<(n4 + BLK - 1) / BLK, BLK, 0, stream>>>((const float4*)pref,
                                                           (float4*)X, n4);
  }
  // 2) X[NU:] = features @ mlp_w^T + mlp_b   (K=128, WMMA f32)
  gemm_bias_wmma<128><<<gemm_blocks_NI, BLK, 0, stream>>>(
      features, mlp_w, mlp_b, X + (size_t)NUSR * 64, NITM);
  // 3) row-wise L2 normalize
  normalize_kernel<<<(NTOT * 32 + BLK - 1) / BLK, BLK, 0, stream>>>(X, NTOT);

  // 4) four GCN layers
  for (int l = 0; l < 4; ++l) {
    const float* cw = (const float*)d_in[6 + 6 * l];
    const float* cb = (const float*)d_in[7 + 6 * l];
    const float* lw = (const float*)d_in[8 + 6 * l];
    const float* lb = (const float*)d_in[9 + 6 * l];
    const float* gw = (const float*)d_in[10 + 6 * l];
    const float* gb = (const float*)d_in[11 + 6 * l];

    // m = x @ conv_w^T + conv_b
    gemm_bias_wmma<64><<<gemm_blocks_N, BLK, 0, stream>>>(X, cw, cb, M, NTOT);
    // h = 0
    {
      int n4 = NTOT * 64 / 4;
      zero4_kernel<<<(n4 + BLK - 1) / BLK, BLK, 0, stream>>>((float4*)H, n4);
    }
    // h[dst] += m[src] over all edges
    {
      long long t = (long long)NEDG * 16;
      scatter_kernel<<<(int)((t + BLK - 1) / BLK), BLK, 0, stream>>>(
          edges, edges + NEDG, M, H, NEDG);
    }
    // x = lrelu(lrelu(h)@g_w^T + g_b + lrelu(x@lin_w^T + lin_b) + id_emb)
    combine_kernel<<<gemm_blocks_N, BLK, 0, stream>>>(X, H, id_emb,
                                                      lw, lb, gw, gb, NTOT);
  }
}